// TransformerEncoder_19447611916536
// MI455X (gfx1250) — compile-verified
//
#include <hip/hip_runtime.h>
#include <hip/hip_bf16.h>
#include <math.h>

// MI455X / gfx1250, wave32. FP32 transformer encoder forward.
// GEMMs use V_WMMA_F32_16X16X4_F32 (exact fp32 MAC via the matrix pipe).
// Tiles staged in LDS as [m][k] / [n][k] with stride 36 floats:
//   - 16B-aligned rows (36*4=144 bytes) -> float4 LDS stores, b64 frag loads
//   - lane stride 36 mod 64 banks -> conflict-free v2f fragment loads

typedef __attribute__((ext_vector_type(2))) float v2f;
typedef __attribute__((ext_vector_type(8))) float v8f;

#define BM 128
#define BN 64
#define BK 32
#define LDK 36  // padded K stride

template <bool BTRANS>
__device__ __forceinline__ void load_tiles(const float* __restrict__ A,
                                           const float* __restrict__ Bm, int K,
                                           int N, int rowBlk, int colBlk,
                                           int k0, int tid, float4 ra[4],
                                           float4 rb[2]) {
#pragma unroll
  for (int i = 0; i < 4; ++i) {  // 128x32 A tile: 1024 float4 / 256 thr
    int idx = tid + i * 256;
    int r = idx >> 3, c4 = idx & 7;
    ra[i] = *(const float4*)(A + (size_t)(rowBlk + r) * K + k0 + c4 * 4);
  }
  if (BTRANS) {  // B(k,n) = Bm[n*K+k] : rows are n, contiguous in k
#pragma unroll
    for (int i = 0; i < 2; ++i) {
      int idx = tid + i * 256;
      int r = idx >> 3, c4 = idx & 7;
      rb[i] = *(const float4*)(Bm + (size_t)(colBlk + r) * K + k0 + c4 * 4);
    }
  } else {  // B(k,n) = Bm[k*N+n] : rows are k, contiguous in n
#pragma unroll
    for (int i = 0; i < 2; ++i) {
      int idx = tid + i * 256;
      int kk2 = idx >> 4, c4 = idx & 15;
      rb[i] = *(const float4*)(Bm + (size_t)(k0 + kk2) * N + colBlk + c4 * 4);
    }
  }
}

template <bool BTRANS>
__device__ __forceinline__ void store_tiles(float (*lA)[LDK], float (*lB)[LDK],
                                            int tid, const float4 ra[4],
                                            const float4 rb[2]) {
#pragma unroll
  for (int i = 0; i < 4; ++i) {
    int idx = tid + i * 256;
    int r = idx >> 3, c4 = idx & 7;
    *(float4*)&lA[r][c4 * 4] = ra[i];
  }
  if (BTRANS) {
#pragma unroll
    for (int i = 0; i < 2; ++i) {
      int idx = tid + i * 256;
      int r = idx >> 3, c4 = idx & 7;
      *(float4*)&lB[r][c4 * 4] = rb[i];
    }
  } else {  // transpose n-contiguous quad into [n][k]
#pragma unroll
    for (int i = 0; i < 2; ++i) {
      int idx = tid + i * 256;
      int kk2 = idx >> 4, c4 = idx & 15;
      lB[c4 * 4 + 0][kk2] = rb[i].x;
      lB[c4 * 4 + 1][kk2] = rb[i].y;
      lB[c4 * 4 + 2][kk2] = rb[i].z;
      lB[c4 * 4 + 3][kk2] = rb[i].w;
    }
  }
}

// C[m,n] = scale * sum_k A[m,k]*B(k,n)  (+ epilogue)
// EPI 0: out = C + (bias ? bias[n] : 0)
// EPI 1: out = resid + C
// EPI 2: out = resid + relu(C + bias[n])
template <int EPI, bool BTRANS>
__global__ __launch_bounds__(256) void gemm_wmma_kernel(
    const float* __restrict__ A, const float* __restrict__ Bm,
    const float* __restrict__ bias, const float* __restrict__ resid,
    float* __restrict__ out, int M, int N, int K, float scale,
    long strideA, long strideB, long strideC, long strideR) {
  const int batch = blockIdx.z;
  A += (size_t)batch * strideA;
  Bm += (size_t)batch * strideB;
  out += (size_t)batch * strideC;
  if (EPI != 0) resid += (size_t)batch * strideR;

  __shared__ float lAs[2][BM][LDK];
  __shared__ float lBs[2][BN][LDK];

  const int tid = threadIdx.x;
  const int lane = tid & 31;
  const int wid = tid >> 5;         // 8 waves
  const int m0w = (wid & 3) * 32;   // 4 waves along M, 32 rows each
  const int n0w = (wid >> 2) * 32;  // 2 waves along N, 32 cols each
  const int kh = lane >> 4;         // lane half: K pair select / M half of C
  const int l15 = lane & 15;

  const int rowBlk = blockIdx.y * BM;
  const int colBlk = blockIdx.x * BN;

  v8f acc[2][2] = {{{}, {}}, {{}, {}}};
  float4 ra[4];
  float4 rb[2];

  const int numKB = K / BK;
  load_tiles<BTRANS>(A, Bm, K, N, rowBlk, colBlk, 0, tid, ra, rb);
  store_tiles<BTRANS>(lAs[0], lBs[0], tid, ra, rb);
  __syncthreads();

  for (int kb = 0; kb < numKB; ++kb) {
    const int cur = kb & 1;
    const bool hasNext = (kb + 1) < numKB;
    if (hasNext)  // issue next block's global loads; overlap with WMMAs below
      load_tiles<BTRANS>(A, Bm, K, N, rowBlk, colBlk, (kb + 1) * BK, tid, ra, rb);

#pragma unroll
    for (int kk = 0; kk < BK; kk += 4) {
      const int ko = kk + 2 * kh;  // lanes 0-15: K={kk,kk+1}; 16-31: {kk+2,kk+3}
      v2f a0 = *(const v2f*)&lAs[cur][m0w + l15][ko];
      v2f a1 = *(const v2f*)&lAs[cur][m0w + 16 + l15][ko];
      v2f b0 = *(const v2f*)&lBs[cur][n0w + l15][ko];
      v2f b1 = *(const v2f*)&lBs[cur][n0w + 16 + l15][ko];
      acc[0][0] = __builtin_amdgcn_wmma_f32_16x16x4_f32(false, a0, false, b0,
                                                        (short)0, acc[0][0],
                                                        false, false);
      acc[0][1] = __builtin_amdgcn_wmma_f32_16x16x4_f32(false, a0, false, b1,
                                                        (short)0, acc[0][1],
                                                        false, false);
      acc[1][0] = __builtin_amdgcn_wmma_f32_16x16x4_f32(false, a1, false, b0,
                                                        (short)0, acc[1][0],
                                                        false, false);
      acc[1][1] = __builtin_amdgcn_wmma_f32_16x16x4_f32(false, a1, false, b1,
                                                        (short)0, acc[1][1],
                                                        false, false);
    }
    if (hasNext)
      store_tiles<BTRANS>(lAs[cur ^ 1], lBs[cur ^ 1], tid, ra, rb);
    __syncthreads();
  }

  // C layout: lane(0-15)=N col, VGPR r -> M=r (lanes 0-15) / M=r+8 (16-31)
  const int col0 = colBlk + n0w + l15;
  const int col1 = col0 + 16;
#pragma unroll
  for (int mi = 0; mi < 2; ++mi) {
#pragma unroll
    for (int r = 0; r < 8; ++r) {
      int row = rowBlk + m0w + mi * 16 + r + 8 * kh;
      float v0 = acc[mi][0][r] * scale;
      float v1 = acc[mi][1][r] * scale;
      if (EPI == 0) {
        if (bias) { v0 += bias[col0]; v1 += bias[col1]; }
      } else if (EPI == 1) {
        v0 += resid[(size_t)row * N + col0];
        v1 += resid[(size_t)row * N + col1];
      } else {
        v0 = resid[(size_t)row * N + col0] + fmaxf(v0 + bias[col0], 0.0f);
        v1 = resid[(size_t)row * N + col1] + fmaxf(v1 + bias[col1], 0.0f);
      }
      out[(size_t)row * N + col0] = v0;
      out[(size_t)row * N + col1] = v1;
    }
  }
}

// x[b,s,:] = embedding[token] + PE(s,:)
__global__ __launch_bounds__(256) void embed_pe_kernel(
    const int* __restrict__ inputs, const float* __restrict__ emb,
    float* __restrict__ x, int S, int H) {
  const int idx = blockIdx.x;  // b*S + s
  const int token = inputs[idx];
  const float s = (float)(idx % S);
  const float negLn1e4 = -9.210340371976184f;  // -ln(10000)
  for (int c = threadIdx.x; c < H; c += 256) {
    float v = emb[(size_t)token * H + c];
    int j = c >> 1;
    float angle = s * expf(negLn1e4 * ((float)(2 * j) / (float)H));
    v += (c & 1) ? cosf(angle) : sinf(angle);
    x[(size_t)idx * H + c] = v;
  }
}

// In-place softmax over rows of length 512 (one block per row, wave32).
__global__ __launch_bounds__(256) void softmax512_kernel(float* __restrict__ sc) {
  float* p = sc + (size_t)blockIdx.x * 512;
  const int tid = threadIdx.x;
  const int wid = tid >> 5, lane = tid & 31;
  __shared__ float red[8];
  __shared__ float bcast[2];

  float x0 = p[tid], x1 = p[tid + 256];

  float m = fmaxf(x0, x1);
#pragma unroll
  for (int off = 16; off > 0; off >>= 1) m = fmaxf(m, __shfl_xor(m, off, 32));
  if (lane == 0) red[wid] = m;
  __syncthreads();
  if (wid == 0) {
    float t = red[lane & 7];
#pragma unroll
    for (int off = 4; off > 0; off >>= 1) t = fmaxf(t, __shfl_xor(t, off, 32));
    if (lane == 0) bcast[0] = t;
  }
  __syncthreads();
  m = bcast[0];

  float e0 = expf(x0 - m), e1 = expf(x1 - m);
  float s = e0 + e1;
#pragma unroll
  for (int off = 16; off > 0; off >>= 1) s += __shfl_xor(s, off, 32);
  if (lane == 0) red[wid] = s;
  __syncthreads();
  if (wid == 0) {
    float t = red[lane & 7];
#pragma unroll
    for (int off = 4; off > 0; off >>= 1) t += __shfl_xor(t, off, 32);
    if (lane == 0) bcast[1] = t;
  }
  __syncthreads();
  float inv = 1.0f / bcast[1];
  p[tid] = e0 * inv;
  p[tid + 256] = e1 * inv;
}

extern "C" void kernel_launch(void* const* d_in, const int* in_sizes, int n_in,
                              void* d_out, int out_size, void* d_ws,
                              size_t ws_size, hipStream_t stream) {
  const int B = 8, S = 512, H = 1024, L = 6;
  const int BS = B * S;  // 4096

  const int* inputs = (const int*)d_in[0];
  const float* emb = (const float*)d_in[1];
  const float* Wq = (const float*)d_in[2];
  const float* bq = (const float*)d_in[3];
  const float* Wk = (const float*)d_in[4];
  const float* bk = (const float*)d_in[5];
  const float* Wv = (const float*)d_in[6];
  const float* bv = (const float*)d_in[7];
  const float* Wm = (const float*)d_in[8];
  const float* bm = (const float*)d_in[9];
  float* out = (float*)d_out;

  // workspace: xA,xB,q,k,v (B*S*H each) + scores (B*S*S) ~= 92.3 MB
  const size_t sz = (size_t)BS * H;
  float* xA = (float*)d_ws;
  float* xB = xA + sz;
  float* q = xB + sz;
  float* k = q + sz;
  float* v = k + sz;
  float* sc = v + sz;  // B*S*S

  const float scale = 0.03125f;  // 1/sqrt(1024)

  embed_pe_kernel<<<dim3(BS), dim3(256), 0, stream>>>(inputs, emb, xA, S, H);

  for (int i = 0; i < L; ++i) {
    const float* Wqi = Wq + (size_t)i * H * H;
    const float* Wki = Wk + (size_t)i * H * H;
    const float* Wvi = Wv + (size_t)i * H * H;
    const float* Wmi = Wm + (size_t)i * H * H;
    const float* bqi = bq + (size_t)i * H;
    const float* bki = bk + (size_t)i * H;
    const float* bvi = bv + (size_t)i * H;
    const float* bmi = bm + (size_t)i * H;

    // q/k/v = x @ W^T + b   (M=4096, N=1024, K=1024)
    dim3 gQKV(H / BN, BS / BM, 1);
    gemm_wmma_kernel<0, true><<<gQKV, 256, 0, stream>>>(
        xA, Wqi, bqi, nullptr, q, BS, H, H, 1.0f, 0, 0, 0, 0);
    gemm_wmma_kernel<0, true><<<gQKV, 256, 0, stream>>>(
        xA, Wki, bki, nullptr, k, BS, H, H, 1.0f, 0, 0, 0, 0);
    gemm_wmma_kernel<0, true><<<gQKV, 256, 0, stream>>>(
        xA, Wvi, bvi, nullptr, v, BS, H, H, 1.0f, 0, 0, 0, 0);

    // scores[b] = scale * q[b] @ k[b]^T   (M=N=512, K=1024, batched over B)
    dim3 gSc(S / BN, S / BM, B);
    gemm_wmma_kernel<0, true><<<gSc, 256, 0, stream>>>(
        q, k, nullptr, nullptr, sc, S, S, H, scale,
        (long)S * H, (long)S * H, (long)S * S, 0);

    softmax512_kernel<<<dim3(BS), dim3(256), 0, stream>>>(sc);

    // xB[b] = xA[b] + attn[b] @ v[b]   (M=512, N=1024, K=512, batched)
    dim3 gCtx(H / BN, S / BM, B);
    gemm_wmma_kernel<1, false><<<gCtx, 256, 0, stream>>>(
        sc, v, nullptr, xA, xB, S, H, S, 1.0f,
        (long)S * S, (long)S * H, (long)S * H, (long)S * H);

    // x_next = xB + relu(xB @ Wm^T + bm)  (last layer writes d_out directly)
    float* dst = (i == L - 1) ? out : xA;
    gemm_wmma_kernel<2, true><<<gQKV, 256, 0, stream>>>(
        xB, Wmi, bmi, xB, dst, BS, H, H, 1.0f, 0, 0, 0, 0);
  }
}